// Actor_midlayer_46875273068974
// MI455X (gfx1250) — compile-verified
//
#include <hip/hip_runtime.h>
#include <math.h>

// Problem constants (match reference).
#define N_NODES 1024
#define N_EDGES 524288
#define FE      64
#define FC      128
#define LAYERS  3
#define NEG_SLOPE 0.2f

// LDS padding strides (floats) chosen for bank-conflict-free ds_load_b64:
//   A rows: 68  -> bank = (4*l16 + 4s + 2hi) % 64, all 32 lanes disjoint
//   B k-pair rows: 160 -> lo half banks [c..c+31], hi half [c+32..c+63]
#define A_STRIDE 68
#define B_STRIDE 160

typedef float v2f __attribute__((ext_vector_type(2)));
typedef float v8f __attribute__((ext_vector_type(8)));

#ifndef __has_builtin
#define __has_builtin(x) 0
#endif

// Async global->LDS copy path: emit the VGLOBAL async instruction via inline
// asm (the clang builtin's pointer params use CUDA LangAS qualifiers that HIP
// source cannot spell), and use the probe-confirmed wait-counter builtin.
#if __has_builtin(__builtin_amdgcn_s_wait_asynccnt)
#define USE_ASYNC_LDS 1
#else
#define USE_ASYNC_LDS 0
#endif

// ---------------------------------------------------------------------------
// float atomic max via int/uint ordering trick (m_buf initialized to -inf)
// ---------------------------------------------------------------------------
__device__ __forceinline__ void atomic_max_f32(float* addr, float val) {
    if (val >= 0.0f) atomicMax((int*)addr, __float_as_int(val));
    else             atomicMin((unsigned int*)addr, __float_as_uint(val));
}

// ---------------------------------------------------------------------------
// d_out <- node_feature   (node state lives in d_out, updated in place)
// ---------------------------------------------------------------------------
__global__ __launch_bounds__(256) void copy_kernel(const float* __restrict__ in,
                                                   float* __restrict__ out) {
    const int i = blockIdx.x * 256 + threadIdx.x;
    out[i] = in[i];
}

// ---------------------------------------------------------------------------
// ef[e][f] = x0*W_in[0][f] + x1*W_in[1][f] + b[f];  thread = (edge, 4-feat chunk)
// ---------------------------------------------------------------------------
__global__ __launch_bounds__(256) void ef_init_kernel(const float* __restrict__ x,
                                                      const float* __restrict__ Win,
                                                      const float* __restrict__ bin,
                                                      float* __restrict__ ef) {
    const int tid = blockIdx.x * 256 + threadIdx.x;      // E*16 total
    const int e   = tid >> 4;
    const int f   = (tid & 15) * 4;
    const float x0 = x[2 * e], x1 = x[2 * e + 1];
    const float4 w0 = *(const float4*)(Win + f);
    const float4 w1 = *(const float4*)(Win + FE + f);
    const float4 b  = *(const float4*)(bin + f);
    float4 o;
    o.x = fmaf(x0, w0.x, fmaf(x1, w1.x, b.x));
    o.y = fmaf(x0, w0.y, fmaf(x1, w1.y, b.y));
    o.z = fmaf(x0, w0.z, fmaf(x1, w1.z, b.z));
    o.w = fmaf(x0, w0.w, fmaf(x1, w1.w, b.w));
    *(float4*)(ef + (size_t)e * FE + f) = o;
}

// ---------------------------------------------------------------------------
// Per-node projections: f_ni = node@W_ni, f_nj = node@W_nj, h = node@W_node + b.
// One block per node; 256 threads = 64 + 64 + 128 output columns exactly.
// ---------------------------------------------------------------------------
__global__ __launch_bounds__(256) void node_proj_kernel(
    const float* __restrict__ node, const float* __restrict__ Wni,
    const float* __restrict__ Wnj,  const float* __restrict__ Wnode,
    const float* __restrict__ bias, float* __restrict__ f_ni,
    float* __restrict__ f_nj, float* __restrict__ h) {
    __shared__ float row[FC];
    const int n = blockIdx.x;
    const int t = threadIdx.x;
    if (t < FC) row[t] = node[n * FC + t];
    __syncthreads();
    if (t < 64) {
        float s = 0.f;
        for (int k = 0; k < FC; ++k) s = fmaf(row[k], Wni[k * FE + t], s);
        f_ni[n * FE + t] = s;
    } else if (t < 128) {
        const int c = t - 64;
        float s = 0.f;
        for (int k = 0; k < FC; ++k) s = fmaf(row[k], Wnj[k * FE + c], s);
        f_nj[n * FE + c] = s;
    } else {
        const int c = t - 128;
        float s = bias[c];
        for (int k = 0; k < FC; ++k) s = fmaf(row[k], Wnode[k * FC + c], s);
        h[n * FC + c] = s;
    }
}

// ---------------------------------------------------------------------------
// Per-layer reset: m = -inf, z = 0, new_node = 0.   Grid covers N*FC elems.
// ---------------------------------------------------------------------------
__global__ __launch_bounds__(256) void init_layer_kernel(float* m_buf, float* z_buf,
                                                         float* nn) {
    const int i = blockIdx.x * 256 + threadIdx.x;
    if (i < N_NODES) { m_buf[i] = -__builtin_inff(); z_buf[i] = 0.f; }
    nn[i] = 0.f;
}

// ---------------------------------------------------------------------------
// Fused edge kernel (WMMA workhorse). Wave = 16-edge tile.
//   - A tile (16x64, contiguous 4KB) copied global->LDS coalesced via the
//     CDNA5 async-to-LDS path (ASYNCcnt) when available.
//   - W_fij staged pair-interleaved in LDS so each B fragment is ONE
//     ds_load_b64 straight into an aligned VGPR pair (no v_mov marshaling).
//   - 64x v_wmma_f32_16x16x4_f32 per wave tile (full fp32 precision;
//     workload is memory-bound at ~10 FLOP/B so fp32 is free).
//   - Fixup: leaky_relu(gemm + f_ni[src] + f_nj[dst]); ef_new = ef_old(LDS) + v
//     (saves the global re-read), e = f_out.attn via 16-lane shfl_xor reduce,
//     atomic float-max into m[dst].
// Fragment layouts per ISA 7.12.2 (f32):
//   A 16x4 : M = lane%16, K = v + 2*(lane/16)
//   B 4x16 : N = lane%16, K = v + 2*(lane/16)
//   C/D    : M = r + 8*(lane/16), N = 16*t + lane%16
// ---------------------------------------------------------------------------
__global__ __launch_bounds__(256) void egat_edge_kernel(
    float* __restrict__ ef,          // [E,64] read + in-place update
    const float* __restrict__ Wfij,  // [64,64] layer slice
    const float* __restrict__ f_ni,  // [N,64]
    const float* __restrict__ f_nj,  // [N,64]
    const float* __restrict__ attn,  // [64]
    const int* __restrict__ src,
    const int* __restrict__ dst,
    float* __restrict__ e_buf,       // [E]
    float* m_buf)                    // [N]
{
    __shared__ float sW[32 * B_STRIDE];        // k-pair interleaved W_fij (20KB)
    __shared__ float sA[8 * 16 * A_STRIDE];    // per-wave padded A tiles (34KB)
    __shared__ int   sSrc[128];
    __shared__ int   sDst[128];

    const int tid  = threadIdx.x;
    const int wave = tid >> 5;
    const int lane = tid & 31;
    const int hi   = lane >> 4;
    const int l16  = lane & 15;
    const int blockEdge0 = blockIdx.x * 128;   // 8 waves x 16 edges

    // Stage B: W[k][n] -> sW[(k>>1)*B_STRIDE + 2n + (k&1)]  (pairs contiguous)
    for (int i = tid; i < FE * FE; i += 256) {
        const int k = i >> 6, n = i & 63;
        sW[(k >> 1) * B_STRIDE + 2 * n + (k & 1)] = Wfij[i];
    }
    if (tid < 128) {
        sSrc[tid] = src[blockEdge0 + tid];
        sDst[tid] = dst[blockEdge0 + tid];
    }

    const int edge0 = blockEdge0 + wave * 16;
    float* sAw = sA + wave * (16 * A_STRIDE);
    const float* gtile = ef + (size_t)edge0 * FE;   // contiguous 4KB tile

    // Stage A tile, coalesced 16B/lane; padded rows in LDS.
#if USE_ASYNC_LDS
    #pragma unroll
    for (int i = 0; i < 8; ++i) {
        const int goff = (lane + i * 32) * 4;           // float index in tile
        const int row = goff >> 6, col = goff & 63;
        const float* gaddr = gtile + goff;
        // Low 32 bits of a flat shared pointer are the LDS byte offset
        // (ISA 10.2: LDS aperture base occupies bits [63:32] only).
        const unsigned ldsaddr =
            (unsigned)(unsigned long long)(sAw + row * A_STRIDE + col);
        asm volatile("global_load_async_to_lds_b128 %0, %1, off"
                     :: "v"(ldsaddr), "v"(gaddr) : "memory");
    }
    __builtin_amdgcn_s_wait_asynccnt(0);
#else
    #pragma unroll
    for (int i = 0; i < 8; ++i) {
        const int goff = (lane + i * 32) * 4;
        const int row = goff >> 6, col = goff & 63;
        const float4 gv = *(const float4*)(gtile + goff);
        *(float4*)(sAw + row * A_STRIDE + col) = gv;
    }
#endif
    __syncthreads();

    v8f zero8 = {0.f, 0.f, 0.f, 0.f, 0.f, 0.f, 0.f, 0.f};
    v8f acc[4] = {zero8, zero8, zero8, zero8};

    #pragma unroll
    for (int s = 0; s < 16; ++s) {            // K = 4*s .. 4*s+3
        const int kb = 4 * s + 2 * hi;
        const v2f a = *(const v2f*)(sAw + l16 * A_STRIDE + kb);       // 1x ds_load_b64
        const float* bprow = sW + (2 * s + hi) * B_STRIDE;
        #pragma unroll
        for (int t = 0; t < 4; ++t) {         // N tiles 0..3
            const v2f b = *(const v2f*)(bprow + 2 * (16 * t + l16));  // 1x ds_load_b64
            acc[t] = __builtin_amdgcn_wmma_f32_16x16x4_f32(
                false, a, false, b, (short)0, acc[t], false, false);
        }
    }

    float attn_r[4];
    #pragma unroll
    for (int t = 0; t < 4; ++t) attn_r[t] = attn[16 * t + l16];

    float rowsum[8];
    #pragma unroll
    for (int r = 0; r < 8; ++r) {
        const int M  = r + 8 * hi;
        const int le = wave * 16 + M;
        const float* ni = f_ni + (size_t)sSrc[le] * FE;   // 256KB tables, cache-hot
        const float* nj = f_nj + (size_t)sDst[le] * FE;
        float* erow = ef + (size_t)(edge0 + M) * FE;
        const float* aold = sAw + M * A_STRIDE;           // old ef row from LDS
        float part = 0.f;
        #pragma unroll
        for (int t = 0; t < 4; ++t) {
            const int n = 16 * t + l16;
            float v = acc[t][r] + ni[n] + nj[n];
            v = (v > 0.f) ? v : NEG_SLOPE * v;            // leaky_relu
            erow[n] = aold[n] + v;                        // ef += f_out (no global re-read)
            part = fmaf(v, attn_r[t], part);
        }
        // reduce over the 16 lanes holding row M (xor 1/2/4/8 stays in half-wave)
        part += __shfl_xor(part, 1);
        part += __shfl_xor(part, 2);
        part += __shfl_xor(part, 4);
        part += __shfl_xor(part, 8);
        rowsum[r] = part;
    }

    if (l16 == 0) {
        #pragma unroll
        for (int r = 0; r < 8; ++r) {
            const int M = r + 8 * hi;
            e_buf[edge0 + M] = rowsum[r];
            atomic_max_f32(m_buf + sDst[wave * 16 + M], rowsum[r]);
        }
    }
}

// ---------------------------------------------------------------------------
// ex = exp(e - m[dst]); store ex in place; z[dst] += ex.
// ---------------------------------------------------------------------------
__global__ __launch_bounds__(256) void softmax_z_kernel(float* e_buf,
                                                        const float* __restrict__ m_buf,
                                                        const int* __restrict__ dst,
                                                        float* z_buf) {
    const int i = blockIdx.x * 256 + threadIdx.x;
    const int d = dst[i];
    const float ex = __expf(e_buf[i] - m_buf[d]);
    e_buf[i] = ex;
    atomicAdd(z_buf + d, ex);
}

// ---------------------------------------------------------------------------
// new_node[dst] += (ex/z[dst]) * h[src].  Thread = (edge, 4-feature chunk).
// ---------------------------------------------------------------------------
__global__ __launch_bounds__(256) void aggregate_kernel(
    const float* __restrict__ aex, const float* __restrict__ z_buf,
    const int* __restrict__ src, const int* __restrict__ dst,
    const float* __restrict__ h, float* nn) {
    const int tid = blockIdx.x * 256 + threadIdx.x;   // E*32 total
    const int e = tid >> 5;
    const int c = tid & 31;
    const int d = dst[e];
    const float a = aex[e] / z_buf[d];
    const float4 hv = *(const float4*)(h + (size_t)src[e] * FC + 4 * c);
    float* o = nn + (size_t)d * FC + 4 * c;
    atomicAdd(o + 0, a * hv.x);
    atomicAdd(o + 1, a * hv.y);
    atomicAdd(o + 2, a * hv.z);
    atomicAdd(o + 3, a * hv.w);
}

// ---------------------------------------------------------------------------
// Residual: node += new_node.
// ---------------------------------------------------------------------------
__global__ __launch_bounds__(256) void node_add_kernel(float* node,
                                                       const float* __restrict__ nn) {
    const int i = blockIdx.x * 256 + threadIdx.x;
    node[i] += nn[i];
}

// ---------------------------------------------------------------------------
// Driver. Workspace layout (floats), total ~137 MB:
//   ef[E*64] | e/ex[E] | m[N] | z[N] | f_ni[N*64] | f_nj[N*64] | h[N*128] | nn[N*128]
// ---------------------------------------------------------------------------
extern "C" void kernel_launch(void* const* d_in, const int* in_sizes, int n_in,
                              void* d_out, int out_size, void* d_ws, size_t ws_size,
                              hipStream_t stream) {
    const float* node_feature = (const float*)d_in[0];
    const float* edge_feature = (const float*)d_in[1];
    const int*   src          = (const int*)d_in[2];
    const int*   dst          = (const int*)d_in[3];
    const float* W_in         = (const float*)d_in[4];
    const float* b_in         = (const float*)d_in[5];
    const float* W_ni         = (const float*)d_in[6];
    const float* W_nj         = (const float*)d_in[7];
    const float* W_fij        = (const float*)d_in[8];
    const float* W_node       = (const float*)d_in[9];
    const float* attn         = (const float*)d_in[10];
    const float* bias_node    = (const float*)d_in[11];
    float* node = (float*)d_out;

    float* ws = (float*)d_ws;
    size_t o = 0;
    float* ef   = ws;      o += (size_t)N_EDGES * FE;
    float* ebuf = ws + o;  o += N_EDGES;
    float* mbuf = ws + o;  o += N_NODES;
    float* zbuf = ws + o;  o += N_NODES;
    float* fni  = ws + o;  o += (size_t)N_NODES * FE;
    float* fnj  = ws + o;  o += (size_t)N_NODES * FE;
    float* hbuf = ws + o;  o += (size_t)N_NODES * FC;
    float* nn   = ws + o;

    copy_kernel<<<(N_NODES * FC) / 256, 256, 0, stream>>>(node_feature, node);
    ef_init_kernel<<<(N_EDGES * 16) / 256, 256, 0, stream>>>(edge_feature, W_in, b_in, ef);

    for (int l = 0; l < LAYERS; ++l) {
        node_proj_kernel<<<N_NODES, 256, 0, stream>>>(
            node,
            W_ni + (size_t)l * FC * FE,
            W_nj + (size_t)l * FC * FE,
            W_node + (size_t)l * FC * FC,
            bias_node + (size_t)l * FC,
            fni, fnj, hbuf);
        init_layer_kernel<<<(N_NODES * FC) / 256, 256, 0, stream>>>(mbuf, zbuf, nn);
        egat_edge_kernel<<<N_EDGES / 128, 256, 0, stream>>>(
            ef, W_fij + (size_t)l * FE * FE, fni, fnj, attn + (size_t)l * FE,
            src, dst, ebuf, mbuf);
        softmax_z_kernel<<<N_EDGES / 256, 256, 0, stream>>>(ebuf, mbuf, dst, zbuf);
        aggregate_kernel<<<(N_EDGES * 32) / 256, 256, 0, stream>>>(
            ebuf, zbuf, src, dst, hbuf, nn);
        node_add_kernel<<<(N_NODES * FC) / 256, 256, 0, stream>>>(node, nn);
    }
}